// DGCNN_Aux_19353122636486
// MI455X (gfx1250) — compile-verified
//
#include <hip/hip_runtime.h>
#include <hip/hip_bf16.h>

typedef float v2f __attribute__((ext_vector_type(2)));
typedef float v8f __attribute__((ext_vector_type(8)));

#define GRAPHS 128
#define NPTS   256
#define KNN    30

__device__ __forceinline__ v8f wmma_f32(v2f a, v2f b, v8f c) {
  // D = A(16x4,f32) * B(4x16,f32) + C(16x16,f32)
  return __builtin_amdgcn_wmma_f32_16x16x4_f32(false, a, false, b, (short)0, c, false, false);
}

// ---------------------------------------------------------------- features
__global__ void build_x0_kernel(const float* __restrict__ pc,
                                const float* __restrict__ fs,
                                float* __restrict__ x0) {
  int t = blockIdx.x * blockDim.x + threadIdx.x;   // 128*256 threads
  if (t >= GRAPHS * NPTS) return;
  int g = t / NPTS;
  float* o = x0 + t * 16;
  const float* p = pc + t * 3;
  const float* f = fs + g * 12;
  o[0] = p[0]; o[1] = p[1]; o[2] = p[2];
#pragma unroll
  for (int c = 0; c < 12; ++c) o[3 + c] = f[c];
  o[15] = 0.f;   // pad so knn/edge GEMMs can run padded widths
}

// ---------------------------------------------------------------- knn
template <int S>
__global__ void knn_kernel(const float* __restrict__ xin, int* __restrict__ nnout) {
  __shared__ float xs[NPTS * S];
  __shared__ float sq[NPTS];
  const int g = blockIdx.x;
  const int i = threadIdx.x;
  const float* xg = xin + (size_t)g * NPTS * S;
  float s = 0.f;
#pragma unroll
  for (int c = 0; c < S; ++c) { float v = xg[i * S + c]; xs[i * S + c] = v; s += v * v; }
  sq[i] = s;
  __syncthreads();
  float myx[S];
#pragma unroll
  for (int c = 0; c < S; ++c) myx[c] = xs[i * S + c];
  float d[NPTS];
  for (int j = 0; j < NPTS; ++j) {
    float dot = 0.f;
#pragma unroll
    for (int c = 0; c < S; ++c) dot += myx[c] * xs[j * S + c];
    d[j] = s + sq[j] - 2.f * dot;
  }
  int* out = nnout + ((size_t)g * NPTS + i) * KNN;
  for (int r = 0; r < KNN; ++r) {             // strict < == top_k tiebreak (lowest idx)
    float best = __builtin_inff(); int bj = 0;
    for (int j = 0; j < NPTS; ++j) if (d[j] < best) { best = d[j]; bj = j; }
    out[r] = bj;
    d[bj] = __builtin_inff();
  }
}

// ---------------------------------------------------------------- edgeconv
// One block = 1 graph x 4 points (1 point per wave). Neighbors padded 30->32
// (slot >=30 repeats neighbor 0; max over k is unaffected).
template <int CIN, int SIN, int K1P>
__global__ void edgeconv_kernel(const float* __restrict__ xin, const int* __restrict__ nn,
    const float* __restrict__ W1, const float* __restrict__ W2, const float* __restrict__ W3,
    const float* __restrict__ B1, const float* __restrict__ B2, const float* __restrict__ B3,
    const float* __restrict__ G1, const float* __restrict__ BB1, const float* __restrict__ M1, const float* __restrict__ V1,
    const float* __restrict__ G2, const float* __restrict__ BB2, const float* __restrict__ M2, const float* __restrict__ V2,
    float* __restrict__ xout) {
  __shared__ float sW1[K1P * 32];
  __shared__ float sW2[32 * 32];
  __shared__ float sW3[32 * 32];
  __shared__ float s1[32], o1[32], s2[32], o2[32], sb3[32];
  __shared__ float stg[4][16 * K1P];
  const int tid = threadIdx.x;
  const int g  = blockIdx.x >> 6;          // /64
  const int p0 = (blockIdx.x & 63) * 4;

  for (int t = tid; t < K1P * 32; t += 128) {
    int k = t >> 5, n = t & 31;
    sW1[t] = (k < 2 * CIN) ? W1[k * 32 + n] : 0.f;   // zero-pad K rows
  }
  for (int t = tid; t < 1024; t += 128) { sW2[t] = W2[t]; sW3[t] = W3[t]; }
  if (tid < 32) {
    float sc1 = G1[tid] * rsqrtf(V1[tid] + 1e-5f);
    s1[tid] = sc1; o1[tid] = sc1 * (B1[tid] - M1[tid]) + BB1[tid];
    float sc2 = G2[tid] * rsqrtf(V2[tid] + 1e-5f);
    s2[tid] = sc2; o2[tid] = sc2 * (B2[tid] - M2[tid]) + BB2[tid];
    sb3[tid] = B3[tid];
  }
  __syncthreads();

  const int wave = tid >> 5, lane = tid & 31;
  const int hi = lane >> 4, lc = lane & 15;
  const int p = p0 + wave;
  const int row = g * NPTS + p;
  const int* nb = nn + (size_t)row * KNN;
  float* tile = stg[wave];
  const float* xi = xin + (size_t)row * SIN;
  float mx0 = -__builtin_inff(), mx1 = -__builtin_inff();

  for (int h = 0; h < 2; ++h) {
    // ---- stage edge features e = [xi, xj - xi] (16 edges x K1P)
    for (int t = lane; t < 16 * K1P; t += 32) {
      int rr = t / K1P, cc = t % K1P;
      int sidx = h * 16 + rr;
      int j = nb[sidx < KNN ? sidx : 0];
      const float* xj = xin + ((size_t)g * NPTS + j) * SIN;
      float val = 0.f;
      if (cc < CIN)          val = xi[cc];
      else if (cc < 2 * CIN) val = xj[cc - CIN] - xi[cc - CIN];
      tile[t] = val;
    }
    __syncthreads();

    // ---- layer 1: (16 x 2CIN) @ (2CIN x 32)
    v8f c0 = {}, c1 = {};
#pragma unroll
    for (int ks = 0; ks < K1P / 4; ++ks) {
      v2f a, b0, b1;
      a[0]  = tile[lc * K1P + ks * 4 + 2 * hi];
      a[1]  = tile[lc * K1P + ks * 4 + 1 + 2 * hi];
      b0[0] = sW1[(ks * 4 + 2 * hi) * 32 + lc];
      b0[1] = sW1[(ks * 4 + 1 + 2 * hi) * 32 + lc];
      b1[0] = sW1[(ks * 4 + 2 * hi) * 32 + 16 + lc];
      b1[1] = sW1[(ks * 4 + 1 + 2 * hi) * 32 + 16 + lc];
      c0 = wmma_f32(a, b0, c0);
      c1 = wmma_f32(a, b1, c1);
    }
    __syncthreads();
#pragma unroll
    for (int v = 0; v < 8; ++v) {
      int r = v + 8 * hi;
      tile[r * 32 + lc]      = fmaxf(s1[lc]      * c0[v] + o1[lc],      0.f);
      tile[r * 32 + 16 + lc] = fmaxf(s1[16 + lc] * c1[v] + o1[16 + lc], 0.f);
    }
    __syncthreads();

    // ---- layer 2: (16 x 32) @ (32 x 32)
    v8f d0 = {}, d1 = {};
#pragma unroll
    for (int ks = 0; ks < 8; ++ks) {
      v2f a, b0, b1;
      a[0]  = tile[lc * 32 + ks * 4 + 2 * hi];
      a[1]  = tile[lc * 32 + ks * 4 + 1 + 2 * hi];
      b0[0] = sW2[(ks * 4 + 2 * hi) * 32 + lc];
      b0[1] = sW2[(ks * 4 + 1 + 2 * hi) * 32 + lc];
      b1[0] = sW2[(ks * 4 + 2 * hi) * 32 + 16 + lc];
      b1[1] = sW2[(ks * 4 + 1 + 2 * hi) * 32 + 16 + lc];
      d0 = wmma_f32(a, b0, d0);
      d1 = wmma_f32(a, b1, d1);
    }
    __syncthreads();
#pragma unroll
    for (int v = 0; v < 8; ++v) {
      int r = v + 8 * hi;
      tile[r * 32 + lc]      = fmaxf(s2[lc]      * d0[v] + o2[lc],      0.f);
      tile[r * 32 + 16 + lc] = fmaxf(s2[16 + lc] * d1[v] + o2[16 + lc], 0.f);
    }
    __syncthreads();

    // ---- layer 3: (16 x 32) @ (32 x 32), bias only, then max over 16 edges
    v8f e0 = {}, e1 = {};
#pragma unroll
    for (int ks = 0; ks < 8; ++ks) {
      v2f a, b0, b1;
      a[0]  = tile[lc * 32 + ks * 4 + 2 * hi];
      a[1]  = tile[lc * 32 + ks * 4 + 1 + 2 * hi];
      b0[0] = sW3[(ks * 4 + 2 * hi) * 32 + lc];
      b0[1] = sW3[(ks * 4 + 1 + 2 * hi) * 32 + lc];
      b1[0] = sW3[(ks * 4 + 2 * hi) * 32 + 16 + lc];
      b1[1] = sW3[(ks * 4 + 1 + 2 * hi) * 32 + 16 + lc];
      e0 = wmma_f32(a, b0, e0);
      e1 = wmma_f32(a, b1, e1);
    }
    __syncthreads();
#pragma unroll
    for (int v = 0; v < 8; ++v) {
      mx0 = fmaxf(mx0, e0[v] + sb3[lc]);
      mx1 = fmaxf(mx1, e1[v] + sb3[16 + lc]);
    }
  }
  // combine the two row-halves (lanes L and L^16 hold same column)
  mx0 = fmaxf(mx0, __shfl_xor(mx0, 16, 32));
  mx1 = fmaxf(mx1, __shfl_xor(mx1, 16, 32));
  if (lane < 16) {
    xout[(size_t)row * 32 + lane]      = mx0;
    xout[(size_t)row * 32 + 16 + lane] = mx1;
  }
}

// ---------------------------------------------------------------- lin1 (+ max over points)
// One block = (graph, 64-col chunk); 4 waves, one 16-col tile each.
__global__ void lin1_max_kernel(const float* __restrict__ x1, const float* __restrict__ x2,
                                const float* __restrict__ x3, const float* __restrict__ W,
                                const float* __restrict__ bias, float* __restrict__ gmax) {
  __shared__ float sW[96 * 64];
  __shared__ float sb[64];
  __shared__ float sA[16 * 96];
  const int tid  = threadIdx.x;
  const int g    = blockIdx.x >> 4;
  const int col0 = (blockIdx.x & 15) * 64;
  for (int t = tid; t < 96 * 64; t += 128) {
    int k = t / 64, n = t % 64;
    sW[t] = W[k * 1024 + col0 + n];
  }
  if (tid < 64) sb[tid] = bias[col0 + tid];
  __syncthreads();
  const int wave = tid >> 5, lane = tid & 31, hi = lane >> 4, lc = lane & 15;
  const int wcol = wave * 16 + lc;
  float mmax = -__builtin_inff();
  for (int mt = 0; mt < 16; ++mt) {
    for (int t = tid; t < 16 * 96; t += 128) {
      int rr = t / 96, cc = t % 96;
      size_t r = (size_t)g * NPTS + mt * 16 + rr;
      float val = (cc < 32) ? x1[r * 32 + cc]
                : (cc < 64) ? x2[r * 32 + cc - 32]
                            : x3[r * 32 + cc - 64];
      sA[t] = val;
    }
    __syncthreads();
    v8f acc = {};
#pragma unroll
    for (int ks = 0; ks < 24; ++ks) {
      v2f a, b;
      a[0] = sA[lc * 96 + ks * 4 + 2 * hi];
      a[1] = sA[lc * 96 + ks * 4 + 1 + 2 * hi];
      b[0] = sW[(ks * 4 + 2 * hi) * 64 + wcol];
      b[1] = sW[(ks * 4 + 1 + 2 * hi) * 64 + wcol];
      acc = wmma_f32(a, b, acc);
    }
#pragma unroll
    for (int v = 0; v < 8; ++v) mmax = fmaxf(mmax, acc[v] + sb[wcol]);
    __syncthreads();
  }
  mmax = fmaxf(mmax, __shfl_xor(mmax, 16, 32));
  if (lane < 16) gmax[(size_t)g * 1024 + col0 + wave * 16 + lane] = mmax;
}

// ---------------------------------------------------------------- temporal mean + head
__global__ void mean_pool_kernel(const float* __restrict__ gmax, float* __restrict__ pooled) {
  int t = blockIdx.x * blockDim.x + threadIdx.x;
  if (t >= 8 * 1024) return;
  int b = t >> 10, c = t & 1023;
  float s = 0.f;
  for (int tt = 0; tt < 16; ++tt) s += gmax[((size_t)(b * 16 + tt) << 10) + c];
  pooled[t] = s * (1.f / 16.f);
}

__global__ void dense_kernel(const float* __restrict__ in, const float* __restrict__ W,
                             const float* __restrict__ b, float* __restrict__ out,
                             int IN, int OUT, int relu) {
  int t = blockIdx.x * blockDim.x + threadIdx.x;
  if (t >= 8 * OUT) return;
  int r = t / OUT, c = t % OUT;
  float s = b[c];
  const float* x = in + (size_t)r * IN;
  for (int k = 0; k < IN; ++k) s += x[k] * W[(size_t)k * OUT + c];
  out[t] = relu ? fmaxf(s, 0.f) : s;
}

// ---------------------------------------------------------------- launch
extern "C" void kernel_launch(void* const* d_in, const int* in_sizes, int n_in,
                              void* d_out, int out_size, void* d_ws, size_t ws_size,
                              hipStream_t stream) {
  const float* pc = (const float*)d_in[0];
  const float* fs = (const float*)d_in[1];

  float* x0 = (float*)d_ws;                       // 128*256*16
  float* x1 = x0 + GRAPHS * NPTS * 16;            // 128*256*32
  float* x2 = x1 + GRAPHS * NPTS * 32;
  float* x3 = x2 + GRAPHS * NPTS * 32;
  int*   nn = (int*)(x3 + GRAPHS * NPTS * 32);    // 128*256*30
  float* gmax   = (float*)(nn + GRAPHS * NPTS * KNN);   // 128*1024
  float* pooled = gmax + GRAPHS * 1024;           // 8*1024
  float* h1 = pooled + 8 * 1024;                  // 8*1024
  float* h2 = h1 + 8 * 1024;                      // 8*256
  float* h3 = h2 + 8 * 256;                       // 8*128

  build_x0_kernel<<<GRAPHS, 256, 0, stream>>>(pc, fs, x0);

#define CONVARGS(b) \
    (const float*)d_in[(b)+0], (const float*)d_in[(b)+1], (const float*)d_in[(b)+2], \
    (const float*)d_in[(b)+3], (const float*)d_in[(b)+4], (const float*)d_in[(b)+5], \
    (const float*)d_in[(b)+6], (const float*)d_in[(b)+7], (const float*)d_in[(b)+8], (const float*)d_in[(b)+9], \
    (const float*)d_in[(b)+10], (const float*)d_in[(b)+11], (const float*)d_in[(b)+12], (const float*)d_in[(b)+13]

  knn_kernel<16><<<GRAPHS, 256, 0, stream>>>(x0, nn);
  edgeconv_kernel<15, 16, 32><<<GRAPHS * 64, 128, 0, stream>>>(x0, nn, CONVARGS(2), x1);

  knn_kernel<32><<<GRAPHS, 256, 0, stream>>>(x1, nn);
  edgeconv_kernel<32, 32, 64><<<GRAPHS * 64, 128, 0, stream>>>(x1, nn, CONVARGS(16), x2);

  knn_kernel<32><<<GRAPHS, 256, 0, stream>>>(x2, nn);
  edgeconv_kernel<32, 32, 64><<<GRAPHS * 64, 128, 0, stream>>>(x2, nn, CONVARGS(30), x3);

  lin1_max_kernel<<<GRAPHS * 16, 128, 0, stream>>>(x1, x2, x3,
      (const float*)d_in[44], (const float*)d_in[45], gmax);

  mean_pool_kernel<<<32, 256, 0, stream>>>(gmax, pooled);

  dense_kernel<<<(8 * 1024 + 255) / 256, 256, 0, stream>>>(pooled,
      (const float*)d_in[46], (const float*)d_in[50], h1, 1024, 1024, 1);
  dense_kernel<<<(8 * 256 + 255) / 256, 256, 0, stream>>>(h1,
      (const float*)d_in[47], (const float*)d_in[51], h2, 1024, 256, 1);
  dense_kernel<<<1, 256, 0, stream>>>(h2,
      (const float*)d_in[48], (const float*)d_in[52], h3, 256, 128, 1);
  dense_kernel<<<1, 256, 0, stream>>>(h3,
      (const float*)d_in[49], (const float*)d_in[53], (float*)d_out, 128, 10, 0);
#undef CONVARGS
}